// MsDroid_18348100289082
// MI455X (gfx1250) — compile-verified
//
#include <hip/hip_runtime.h>
#include <hip/hip_bf16.h>
#include <math.h>

// Problem constants (match reference)
#define NN 10000
#define EE 640000
#define DD 128
#define HH 128
#define CC 2
#define GG 64
#define BN_EPS 1e-5f

typedef float v2f __attribute__((ext_vector_type(2)));
typedef float v8f __attribute__((ext_vector_type(8)));

// ---------------------------------------------------------------------------
// Generic f32 WMMA GEMM: C[M x Nd] = A[M x K] @ B[K x Nd] (+bias, +relu)
// Requirements: M % 16 == 0, Nd == 128 (8 waves/block), K % 4 == 0.
// One wave32 computes one 16x16 tile via chained V_WMMA_F32_16X16X4_F32.
// No divergence before WMMA -> EXEC all-1s as required by ISA 7.12.
// ---------------------------------------------------------------------------
__global__ void wmma_gemm_f32(const float* __restrict__ A,
                              const float* __restrict__ B,
                              float* __restrict__ C,
                              int M, int Nd, int K,
                              const float* __restrict__ bias, int relu) {
  const int lane  = threadIdx.x & 31;
  const int wave  = threadIdx.x >> 5;          // 0..7 -> n tile
  const int m0    = blockIdx.x * 16;
  const int n0    = wave * 16;
  const int mr    = lane & 15;                 // M row for A, N col for B/C
  const int kh    = (lane >> 4) << 1;          // K pair: 0 or 2

  v8f acc = {};                                // f32 accumulator (8 VGPRs)
  for (int k = 0; k < K; k += 4) {
    v2f a, b;
    a.x = A[(m0 + mr) * K + k + kh];
    a.y = A[(m0 + mr) * K + k + kh + 1];
    b.x = B[(k + kh) * Nd + n0 + mr];
    b.y = B[(k + kh + 1) * Nd + n0 + mr];
    acc = __builtin_amdgcn_wmma_f32_16x16x4_f32(
        /*neg_a=*/false, a, /*neg_b=*/false, b,
        /*c_mod=*/(short)0, acc, /*reuse_a=*/false, /*reuse_b=*/false);
  }

  const int col   = n0 + mr;
  const int rbase = (lane >> 4) << 3;          // rows r+0..7 or r+8..15
  const float bv  = bias ? bias[col] : 0.0f;
#pragma unroll
  for (int r = 0; r < 8; ++r) {
    float v = acc[r] + bv;
    if (relu) v = fmaxf(v, 0.0f);
    C[(m0 + rbase + r) * Nd + col] = v;
  }
}

// ---------------------------------------------------------------------------
// Utility / graph kernels
// ---------------------------------------------------------------------------
__global__ void zero_f(float* __restrict__ p, int n) {
  int i = blockIdx.x * blockDim.x + threadIdx.x;
  if (i < n) p[i] = 0.0f;
}

__global__ void degree_k(const int* __restrict__ dst, float* __restrict__ deg, int E) {
  int e = blockIdx.x * blockDim.x + threadIdx.x;
  if (e < E) atomicAdd(&deg[dst[e]], 1.0f);
}

__global__ void dinv_k(const float* __restrict__ deg, float* __restrict__ dinv, int N) {
  int n = blockIdx.x * blockDim.x + threadIdx.x;
  if (n < N) dinv[n] = rsqrtf(deg[n] + 1.0f);  // +1 self loop
}

__global__ void counts_k(const int* __restrict__ batch, float* __restrict__ counts, int N) {
  int n = blockIdx.x * blockDim.x + threadIdx.x;
  if (n < N) atomicAdd(&counts[batch[n]], 1.0f);
}

// out[n,:] = hw[n,:] * dinv[n]^2 + bias  (self-loop + bias); float4 per thread
__global__ void self_bias_k(const float* __restrict__ hw, const float* __restrict__ dinv,
                            const float* __restrict__ bias, float* __restrict__ out, int N) {
  int idx = blockIdx.x * blockDim.x + threadIdx.x;   // over N * H/4
  if (idx >= N * (HH / 4)) return;
  int n = idx >> 5, q = idx & 31;                    // q: float4 index within row
  float di = dinv[n];
  float di2 = di * di;
  float4 v = ((const float4*)hw)[idx];
  float4 bb = ((const float4*)bias)[q];
  float4 o;
  o.x = v.x * di2 + bb.x; o.y = v.y * di2 + bb.y;
  o.z = v.z * di2 + bb.z; o.w = v.w * di2 + bb.w;
  ((float4*)out)[idx] = o;
}

// out[dst,:] += hw[src,:] * dinv[src]*dinv[dst]
// One wave32 per edge; each lane handles 4 channels via one b128 load.
__global__ void scatter_k(const float* __restrict__ hw, const int* __restrict__ src,
                          const int* __restrict__ dst, const float* __restrict__ dinv,
                          float* __restrict__ out) {
  long long tid = (long long)blockIdx.x * blockDim.x + threadIdx.x;
  if (tid >= (long long)EE * 32) return;
  int e    = (int)(tid >> 5);
  int lane = (int)(tid & 31);
  int s = src[e], d = dst[e];
  float norm = dinv[s] * dinv[d];
  float4 v = ((const float4*)(hw + s * HH))[lane];   // global_load_b128, coalesced
  float* ob = out + d * HH + lane * 4;
  atomicAdd(ob + 0, v.x * norm);
  atomicAdd(ob + 1, v.y * norm);
  atomicAdd(ob + 2, v.z * norm);
  atomicAdd(ob + 3, v.w * norm);
}

// in-place relu; optional per-channel sum/sumsq accumulation (BN stats).
// 128 threads = one channel each; register accumulation, 1 atomic/channel/block.
__global__ void relu_stats_k(float* __restrict__ h, float* __restrict__ bnsum,
                             float* __restrict__ bnsq, int N, int rowsPerBlock) {
  int c = threadIdx.x;
  int row0 = blockIdx.x * rowsPerBlock;
  float s = 0.0f, sq = 0.0f;
  for (int r = 0; r < rowsPerBlock; ++r) {
    int n = row0 + r;
    if (n >= N) break;
    float v = fmaxf(h[n * HH + c], 0.0f);
    h[n * HH + c] = v;
    s += v; sq += v * v;
  }
  if (bnsum) { atomicAdd(&bnsum[c], s); atomicAdd(&bnsq[c], sq); }
}

__global__ void bn_apply_k(float* __restrict__ h, const float* __restrict__ bnsum,
                           const float* __restrict__ bnsq, const float* __restrict__ g,
                           const float* __restrict__ b, int N) {
  int idx = blockIdx.x * blockDim.x + threadIdx.x;   // over N * H/4
  if (idx >= N * (HH / 4)) return;
  int q = idx & 31;
  float inv = 1.0f / (float)N;
  float4 sm = ((const float4*)bnsum)[q];
  float4 sv = ((const float4*)bnsq)[q];
  float4 gg = ((const float4*)g)[q];
  float4 bb = ((const float4*)b)[q];
  float4 v = ((const float4*)h)[idx];
  float mu, var;
  mu = sm.x * inv; var = sv.x * inv - mu * mu; v.x = (v.x - mu) * rsqrtf(var + BN_EPS) * gg.x + bb.x;
  mu = sm.y * inv; var = sv.y * inv - mu * mu; v.y = (v.y - mu) * rsqrtf(var + BN_EPS) * gg.y + bb.y;
  mu = sm.z * inv; var = sv.z * inv - mu * mu; v.z = (v.z - mu) * rsqrtf(var + BN_EPS) * gg.z + bb.z;
  mu = sm.w * inv; var = sv.w * inv - mu * mu; v.w = (v.w - mu) * rsqrtf(var + BN_EPS) * gg.w + bb.w;
  ((float4*)h)[idx] = v;
}

// mean-sum via f32 atomics; max via uint-bit atomicMax (valid: values >= 0 post-relu)
__global__ void pool_k(const float* __restrict__ h, const int* __restrict__ batch,
                       float* __restrict__ psum, unsigned* __restrict__ pmax, int N) {
  int idx = blockIdx.x * blockDim.x + threadIdx.x;   // over N * H/4
  if (idx >= N * (HH / 4)) return;
  int n = idx >> 5, q = idx & 31;
  int g = batch[n];
  float4 v = ((const float4*)h)[idx];
  float* ps = psum + g * HH + q * 4;
  unsigned* pm = pmax + g * HH + q * 4;
  atomicAdd(ps + 0, v.x); atomicMax(pm + 0, __float_as_uint(v.x));
  atomicAdd(ps + 1, v.y); atomicMax(pm + 1, __float_as_uint(v.y));
  atomicAdd(ps + 2, v.z); atomicMax(pm + 2, __float_as_uint(v.z));
  atomicAdd(ps + 3, v.w); atomicMax(pm + 3, __float_as_uint(v.w));
}

__global__ void graphx_k(const float* __restrict__ psum, const unsigned* __restrict__ pmax,
                         const float* __restrict__ counts, float* __restrict__ gx) {
  int idx = blockIdx.x * blockDim.x + threadIdx.x;   // G * 2H
  if (idx >= GG * 2 * HH) return;
  int g = idx >> 8, c = idx & 255;
  if (c < HH) gx[idx] = psum[g * HH + c] / counts[g];
  else        gx[idx] = __uint_as_float(pmax[g * HH + (c - HH)]);
}

// fc2 + log_softmax + NLL loss + per-class min/argmin over 64 graphs
__global__ void final_k(const float* __restrict__ hidden, const float* __restrict__ w2,
                        const float* __restrict__ b2, const int* __restrict__ y,
                        float* __restrict__ out) {
  __shared__ float lp0[GG], lp1[GG];
  int g = threadIdx.x;                 // 64 threads
  float z0 = b2[0], z1 = b2[1];
  for (int k = 0; k < HH; ++k) {
    float hv = hidden[g * HH + k];
    z0 += hv * w2[k * CC + 0];
    z1 += hv * w2[k * CC + 1];
  }
  float mx  = fmaxf(z0, z1);
  float lse = mx + logf(expf(z0 - mx) + expf(z1 - mx));
  lp0[g] = z0 - lse;
  lp1[g] = z1 - lse;
  __syncthreads();
  if (g == 0) {
    float loss = 0.0f, mn0 = 1e30f, mn1 = 1e30f;
    int i0 = 0, i1 = 0;
    for (int j = 0; j < GG; ++j) {
      loss -= (y[j] == 0) ? lp0[j] : lp1[j];
      float s0 = expf(lp0[j]), s1 = expf(lp1[j]);
      if (s0 < mn0) { mn0 = s0; i0 = j; }   // strict < -> first occurrence (jnp.argmin)
      if (s1 < mn1) { mn1 = s1; i1 = j; }
    }
    out[0] = mn0; out[1] = mn1;
    out[2] = (float)i0; out[3] = (float)i1;
    out[4] = loss / (float)GG;
  }
}

// ---------------------------------------------------------------------------
extern "C" void kernel_launch(void* const* d_in, const int* in_sizes, int n_in,
                              void* d_out, int out_size, void* d_ws, size_t ws_size,
                              hipStream_t stream) {
  const float* x     = (const float*)d_in[0];
  const int* src     = (const int*)d_in[1];          // edge_index[0]
  const int* dst     = src + EE;                     // edge_index[1]
  const int* batch   = (const int*)d_in[2];
  const int* y       = (const int*)d_in[3];
  const float* Wc[3] = {(const float*)d_in[4], (const float*)d_in[6], (const float*)d_in[8]};
  const float* bc[3] = {(const float*)d_in[5], (const float*)d_in[7], (const float*)d_in[9]};
  const float* bng[2] = {(const float*)d_in[10], (const float*)d_in[12]};
  const float* bnb[2] = {(const float*)d_in[11], (const float*)d_in[13]};
  const float* fc1_w = (const float*)d_in[14];
  const float* fc1_b = (const float*)d_in[15];
  const float* fc2_w = (const float*)d_in[16];
  const float* fc2_b = (const float*)d_in[17];
  float* out = (float*)d_out;

  // workspace carve-out (~15.6 MB of f32); big buffers first -> 16B alignment
  float* ws     = (float*)d_ws;
  float* hw     = ws;                   // N*H
  float* bufA   = hw + NN * HH;         // N*H
  float* bufB   = bufA + NN * HH;       // N*H
  float* psum   = bufB + NN * HH;       // G*H (16B-aligned)
  unsigned* pmax = (unsigned*)(psum + GG * HH);  // G*H (contiguous with psum)
  float* gx     = (float*)pmax + GG * HH;        // G*2H
  float* hidden = gx + GG * 2 * HH;              // G*H
  float* bnsum  = hidden + GG * HH;     // H (16B-aligned)
  float* bnsq   = bnsum + HH;           // H (contiguous with bnsum)
  float* deg    = bnsq + HH;            // N
  float* dinv   = deg + NN;             // N
  float* counts = dinv + NN;            // G

  const int elemNH  = NN * HH;                   // 1,280,000
  const int elemNH4 = NN * (HH / 4);             //   320,000
  const long long scatterT = (long long)EE * 32; // 20,480,000

  // ---- degree / dinv / counts (edge-structure, recomputed each call) ----
  zero_f<<<(NN + 255) / 256, 256, 0, stream>>>(deg, NN);
  zero_f<<<1, 64, 0, stream>>>(counts, GG);
  degree_k<<<(EE + 255) / 256, 256, 0, stream>>>(dst, deg, EE);
  dinv_k<<<(NN + 255) / 256, 256, 0, stream>>>(deg, dinv, NN);
  counts_k<<<(NN + 255) / 256, 256, 0, stream>>>(batch, counts, NN);

  // ---- 3 GCN layers ----
  const float* h = x;
  float* outbuf[3] = {bufA, bufB, bufA};
  for (int l = 0; l < 3; ++l) {
    // hw = h @ W  (WMMA): 625 M-tiles, 8 waves/block over 128 cols
    wmma_gemm_f32<<<NN / 16, 256, 0, stream>>>(h, Wc[l], hw, NN, HH, HH, nullptr, 0);
    float* ob = outbuf[l];
    self_bias_k<<<(elemNH4 + 255) / 256, 256, 0, stream>>>(hw, dinv, bc[l], ob, NN);
    scatter_k<<<(int)((scatterT + 255) / 256), 256, 0, stream>>>(hw, src, dst, dinv, ob);
    if (l < 2) {
      zero_f<<<1, 256, 0, stream>>>(bnsum, 2 * HH);  // bnsum + bnsq contiguous
      relu_stats_k<<<(NN + 63) / 64, HH, 0, stream>>>(ob, bnsum, bnsq, NN, 64);
      bn_apply_k<<<(elemNH4 + 255) / 256, 256, 0, stream>>>(ob, bnsum, bnsq, bng[l], bnb[l], NN);
    } else {
      relu_stats_k<<<(NN + 63) / 64, HH, 0, stream>>>(ob, nullptr, nullptr, NN, 64);
    }
    h = ob;
  }

  // ---- pooling -> graph features ----
  zero_f<<<(2 * GG * HH + 255) / 256, 256, 0, stream>>>(psum, 2 * GG * HH); // psum + pmax (0u == +0.0f)
  pool_k<<<(elemNH4 + 255) / 256, 256, 0, stream>>>(h, batch, psum, pmax, NN);
  graphx_k<<<(GG * 2 * HH + 255) / 256, 256, 0, stream>>>(psum, pmax, counts, gx);

  // ---- fc1 (WMMA, bias + relu fused): [64 x 256] @ [256 x 128] ----
  wmma_gemm_f32<<<GG / 16, 256, 0, stream>>>(gx, fc1_w, hidden, GG, HH, 2 * HH, fc1_b, 1);

  // ---- fc2 + log_softmax + loss + min/argmin ----
  final_k<<<1, GG, 0, stream>>>(hidden, fc2_w, fc2_b, y, out);
}